// IncrementalSelfAttention_146028888391
// MI455X (gfx1250) — compile-verified
//
#include <hip/hip_runtime.h>

typedef __attribute__((ext_vector_type(2))) float v2f;
typedef __attribute__((ext_vector_type(8))) float v8f;

namespace {
constexpr int kB  = 16;
constexpr int kS  = 4096;
constexpr int kS1 = 4097;   // S + 1 new token
constexpr int kSP = 4104;   // padded attn stride (multiple of 8, pad zeroed)
constexpr int kD  = 1024;
constexpr int kH  = 16;
constexpr int kHD = 64;
constexpr float kScale = 0.125f;  // hd^-0.5 = 64^-0.5
}

// ---------------------------------------------------------------------------
// Kernel 1: comb_key = [past_key, key], comb_value = [past_value, value].
// Pure bandwidth: one block per (row, batch, k/v), 256 threads x float4 = 4KB.
// ---------------------------------------------------------------------------
__global__ void concat_kernel(const float* __restrict__ key,
                              const float* __restrict__ value,
                              const float* __restrict__ past_key,
                              const float* __restrict__ past_value,
                              float* __restrict__ combK,
                              float* __restrict__ combV) {
  const int j  = blockIdx.x;
  const int b  = blockIdx.y;
  const int kv = blockIdx.z;
  const float4* src;
  if (j < kS) {
    const float* p = kv ? past_value : past_key;
    src = (const float4*)(p + ((size_t)b * kS + j) * kD);
  } else {
    const float* p = kv ? value : key;
    src = (const float4*)(p + (size_t)b * kD);
  }
  float4* dst = (float4*)((kv ? combV : combK) + ((size_t)b * kS1 + j) * kD);
  dst[threadIdx.x] = src[threadIdx.x];
}

// ---------------------------------------------------------------------------
// Kernel 2: qs[b,i] = (query[b,:] . wq[i,:] + bq[i]) * hd^-0.5   (tiny GEMV)
// ---------------------------------------------------------------------------
__global__ void qproj_kernel(const float* __restrict__ query,
                             const float* __restrict__ w_in,
                             const float* __restrict__ b_in,
                             float* __restrict__ qs) {
  const int idx = blockIdx.x * blockDim.x + threadIdx.x;  // b*D + i
  const int b = idx >> 10;
  const int i = idx & (kD - 1);
  const float4* q4 = (const float4*)(query + (size_t)b * kD);
  const float4* w4 = (const float4*)(w_in + (size_t)i * kD);
  float acc = 0.f;
  for (int e = 0; e < kD / 4; ++e) {
    float4 q = q4[e], w = w4[e];
    acc += q.x * w.x + q.y * w.y + q.z * w.z + q.w * w.w;
  }
  qs[idx] = (acc + b_in[i]) * kScale;
}

// ---------------------------------------------------------------------------
// Kernel 3: u[b,h,e] = sum_d qs[b, h*64+d] * wk[h*64+d, e]   (Wk folded into q)
// ---------------------------------------------------------------------------
__global__ void u_kernel(const float* __restrict__ qs,
                         const float* __restrict__ w_in,
                         float* __restrict__ u) {
  const int idx = blockIdx.x * blockDim.x + threadIdx.x;  // (b*H + h)*D + e
  const int e = idx & (kD - 1);
  const int h = (idx >> 10) & (kH - 1);
  const int b = idx >> 14;
  const float* q = qs + (size_t)b * kD + h * kHD;
  const float* w = w_in + ((size_t)kD + (size_t)h * kHD) * kD + e;  // wk slice
  float acc = 0.f;
#pragma unroll 8
  for (int d = 0; d < kHD; ++d) acc += q[d] * w[(size_t)d * kD];
  u[idx] = acc;
}

// ---------------------------------------------------------------------------
// Kernel 4: scores[b,h,j] = u[b,h,:] . comb_key[b,j,:]  via V_WMMA_F32_16X16X4_F32
// One wave = one 16(head) x 16(key) tile.  Branch-free inner loop:
//  - tail key rows are CLAMPED to a valid row; the resulting garbage only
//    lands in output columns that are never stored (N-dim is safe to clamp).
//  - two accumulator chains (even/odd K) hide WMMA D->C latency.
// A (16x4 f32): lanes 0-15 hold K={0,1}, lanes 16-31 hold K={2,3}  (ISA layout)
// B (4x16 f32): lane = N, same K split across wave halves.
// ---------------------------------------------------------------------------
__global__ void scores_kernel(const float* __restrict__ u,
                              const float* __restrict__ combK,
                              float* __restrict__ scores) {
  const int lane = threadIdx.x & 31;
  const int wave = threadIdx.x >> 5;
  const int b = blockIdx.y;
  const int j0 = (blockIdx.x * 8 + wave) * 16;
  if (j0 >= kS1) return;  // wave-uniform exit; live waves keep EXEC all-1s

  const int n = lane & 15;             // head index (A) / key index (B)
  const int koff = (lane >> 4) << 1;   // 0 for lanes 0-15, 2 for lanes 16-31
  const int row = j0 + n;
  const int rowc = row < kS1 ? row : kS1 - 1;  // clamp: discarded column only
  const float* xr = combK + ((size_t)b * kS1 + rowc) * kD + koff;
  const float* ur = u + ((size_t)b * kH + n) * kD + koff;

  v8f acc0 = {}, acc1 = {};
#pragma unroll 2
  for (int kk = 0; kk < kD; kk += 8) {
    v2f a0 = *(const v2f*)(ur + kk);
    v2f a1 = *(const v2f*)(ur + kk + 4);
    v2f b0 = *(const v2f*)(xr + kk);
    v2f b1 = *(const v2f*)(xr + kk + 4);
    acc0 = __builtin_amdgcn_wmma_f32_16x16x4_f32(false, a0, false, b0,
                                                 (short)0, acc0, false, false);
    acc1 = __builtin_amdgcn_wmma_f32_16x16x4_f32(false, a1, false, b1,
                                                 (short)0, acc1, false, false);
  }
  v8f acc = acc0 + acc1;

  // C/D layout: lanes 0-15 -> M=vgpr, lanes 16-31 -> M=vgpr+8; N = lane&15
  const int hbase = (lane >> 4) << 3;
  const int j = j0 + n;
  if (j < kS1) {
#pragma unroll
    for (int v = 0; v < 8; ++v)
      scores[(size_t)(b * kH + hbase + v) * kSP + j] = acc[v];
  }
}

// ---------------------------------------------------------------------------
// Kernel 5: in-place softmax over j per (b,h) row of the padded scores buffer;
// zero the pad entries [kS1, kSP) so the value-pass WMMA K-tail contributes 0.
// (bk bias dropped: constant per row, softmax-invariant.)
// ---------------------------------------------------------------------------
__global__ void softmax_kernel(float* __restrict__ scores) {
  const int bh = blockIdx.x;  // b*H + h
  float* row = scores + (size_t)bh * kSP;
  __shared__ float red[256];

  float lmax = -3.4e38f;
  for (int j = threadIdx.x; j < kS1; j += 256) lmax = fmaxf(lmax, row[j]);
  red[threadIdx.x] = lmax;
  __syncthreads();
  for (int s = 128; s > 0; s >>= 1) {
    if (threadIdx.x < s) red[threadIdx.x] = fmaxf(red[threadIdx.x], red[threadIdx.x + s]);
    __syncthreads();
  }
  const float m = red[0];
  __syncthreads();

  float lsum = 0.f;
  for (int j = threadIdx.x; j < kS1; j += 256) {
    float e = __expf(row[j] - m);
    row[j] = e;
    lsum += e;
  }
  red[threadIdx.x] = lsum;
  __syncthreads();
  for (int s = 128; s > 0; s >>= 1) {
    if (threadIdx.x < s) red[threadIdx.x] += red[threadIdx.x + s];
    __syncthreads();
  }
  const float inv = 1.f / red[0];
  for (int j = threadIdx.x; j < kS1; j += 256) row[j] *= inv;
  for (int j = kS1 + threadIdx.x; j < kSP; j += 256) row[j] = 0.f;  // zero pad
}

// ---------------------------------------------------------------------------
// Kernel 6: y[b,h,e] = sum_j attn[b,h,j] * comb_value[b,j,e]  via WMMA.
// M=16 heads, N=16 value columns per wave, K runs over the padded j range.
// K-tail rows of comb_value are CLAMPED (valid memory); the matching attn
// A-fragments are exactly zero (softmax pad), so they contribute 0.
// ---------------------------------------------------------------------------
__global__ void y_kernel(const float* __restrict__ attn,
                         const float* __restrict__ combV,
                         float* __restrict__ y) {
  const int lane = threadIdx.x & 31;
  const int wave = threadIdx.x >> 5;
  const int b = blockIdx.y;
  const int e0 = (blockIdx.x * 8 + wave) * 16;  // value-column tile

  const int n = lane & 15;
  const int koff = (lane >> 4) << 1;
  const float* ar = attn + ((size_t)b * kH + n) * kSP + koff;   // A: head = n
  const float* vb = combV + (size_t)b * kS1 * kD + e0 + n;      // B: column e0+n

  v8f acc0 = {}, acc1 = {};
#pragma unroll 2
  for (int j0 = 0; j0 < kSP; j0 += 8) {
    v2f a0 = *(const v2f*)(ar + j0);
    v2f a1 = *(const v2f*)(ar + j0 + 4);
    const int r0 = j0 + koff;
    const int r0c = r0     < kS1 ? r0     : kS1 - 1;  // clamped K rows; attn
    const int r1c = r0 + 1 < kS1 ? r0 + 1 : kS1 - 1;  // pad zeros nullify them
    const int r2c = r0 + 4 < kS1 ? r0 + 4 : kS1 - 1;
    const int r3c = r0 + 5 < kS1 ? r0 + 5 : kS1 - 1;
    v2f b0, b1;
    b0.x = vb[(size_t)r0c * kD];
    b0.y = vb[(size_t)r1c * kD];
    b1.x = vb[(size_t)r2c * kD];
    b1.y = vb[(size_t)r3c * kD];
    acc0 = __builtin_amdgcn_wmma_f32_16x16x4_f32(false, a0, false, b0,
                                                 (short)0, acc0, false, false);
    acc1 = __builtin_amdgcn_wmma_f32_16x16x4_f32(false, a1, false, b1,
                                                 (short)0, acc1, false, false);
  }
  v8f acc = acc0 + acc1;

  const int hbase = (lane >> 4) << 3;
#pragma unroll
  for (int v = 0; v < 8; ++v)
    y[((size_t)b * kH + hbase + v) * kD + e0 + n] = acc[v];
}

// ---------------------------------------------------------------------------
// Kernel 7: ctx[b, h*64+d] = wv[h*64+d,:] . y[b,h,:] + bv   (sum(attn)==1)
// ---------------------------------------------------------------------------
__global__ void ctx_kernel(const float* __restrict__ y,
                           const float* __restrict__ w_in,
                           const float* __restrict__ b_in,
                           float* __restrict__ ctx) {
  const int idx = blockIdx.x * blockDim.x + threadIdx.x;  // b*D + i
  const int b = idx >> 10;
  const int i = idx & (kD - 1);
  const int h = i >> 6;
  const float4* w4 = (const float4*)(w_in + ((size_t)2 * kD + i) * kD);
  const float4* y4 = (const float4*)(y + ((size_t)b * kH + h) * kD);
  float acc = 0.f;
  for (int e = 0; e < kD / 4; ++e) {
    float4 w = w4[e], v = y4[e];
    acc += w.x * v.x + w.y * v.y + w.z * v.z + w.w * v.w;
  }
  ctx[idx] = acc + b_in[2 * kD + i];
}

// ---------------------------------------------------------------------------
// Kernel 8: out[b,i] = w_out[i,:] . ctx[b,:] + b_out[i]
// ---------------------------------------------------------------------------
__global__ void out_kernel(const float* __restrict__ ctx,
                           const float* __restrict__ w_out,
                           const float* __restrict__ b_out,
                           float* __restrict__ out) {
  const int idx = blockIdx.x * blockDim.x + threadIdx.x;  // b*D + i
  const int b = idx >> 10;
  const int i = idx & (kD - 1);
  const float4* w4 = (const float4*)(w_out + (size_t)i * kD);
  const float4* c4 = (const float4*)(ctx + (size_t)b * kD);
  float acc = 0.f;
  for (int e = 0; e < kD / 4; ++e) {
    float4 w = w4[e], c = c4[e];
    acc += w.x * c.x + w.y * c.y + w.z * c.z + w.w * c.w;
  }
  out[idx] = acc + b_out[i];
}

extern "C" void kernel_launch(void* const* d_in, const int* in_sizes, int n_in,
                              void* d_out, int out_size, void* d_ws, size_t ws_size,
                              hipStream_t stream) {
  (void)in_sizes; (void)n_in; (void)out_size; (void)ws_size;
  const float* query      = (const float*)d_in[0];
  const float* key        = (const float*)d_in[1];
  const float* value      = (const float*)d_in[2];
  const float* past_key   = (const float*)d_in[3];
  const float* past_value = (const float*)d_in[4];
  const float* w_in       = (const float*)d_in[5];
  const float* b_in       = (const float*)d_in[6];
  const float* w_out      = (const float*)d_in[7];
  const float* b_out      = (const float*)d_in[8];
  // d_in[9] = nhead (16), hardcoded.

  // Output tuple layout: out (B*D), comb_key (B*S1*D), comb_value (B*S1*D).
  float* out   = (float*)d_out;
  float* combK = out + (size_t)kB * kD;
  float* combV = combK + (size_t)kB * kS1 * kD;

  // Workspace (~6.4 MB of f32).
  float* ws     = (float*)d_ws;
  float* qs     = ws;                                 // B*D
  float* u      = qs + (size_t)kB * kD;               // B*H*D
  float* scores = u + (size_t)kB * kH * kD;           // B*H*kSP (attn in place)
  float* y      = scores + (size_t)kB * kH * kSP;     // B*H*D
  float* ctx    = y + (size_t)kB * kH * kD;           // B*D

  concat_kernel<<<dim3(kS1, kB, 2), 256, 0, stream>>>(key, value, past_key,
                                                      past_value, combK, combV);
  qproj_kernel<<<kB * kD / 256, 256, 0, stream>>>(query, w_in, b_in, qs);
  u_kernel<<<kB * kH * kD / 256, 256, 0, stream>>>(qs, w_in, u);
  scores_kernel<<<dim3((kS1 + 127) / 128, kB), 256, 0, stream>>>(u, combK, scores);
  softmax_kernel<<<kB * kH, 256, 0, stream>>>(scores);
  y_kernel<<<dim3(kD / 128 / 2, kB), 256, 0, stream>>>(scores, combV, y);
  ctx_kernel<<<kB * kD / 256, 256, 0, stream>>>(y, w_in, b_in, ctx);
  out_kernel<<<kB * kD / 256, 256, 0, stream>>>(ctx, w_out, b_out, out);
}